// SetAbstract_51170240364929
// MI455X (gfx1250) — compile-verified
//
#include <hip/hip_runtime.h>

typedef __attribute__((ext_vector_type(2))) float v2f;
typedef __attribute__((ext_vector_type(8))) float v8f;

#define B_    4
#define N_    8192
#define D_    32
#define M0_   32
#define M1_   32
#define M2_   64
#define KNN_  16
#define LEAKY_ 0.1f
#define W0COLS (D_ + 3)   // 35 columns: 3 xyz + 32 feature

__device__ __forceinline__ float lrelu(float v) { return fmaxf(v, LEAKY_ * v); }

__device__ __forceinline__ v8f wmma_f32_4(v2f a, v2f b, v8f c) {
  return __builtin_amdgcn_wmma_f32_16x16x4_f32(false, a, false, b, (short)0, c,
                                               false, false);
}

// ---------------------------------------------------------------------------
// Kernel A: feat[b][n][o] = sum_c W0[o][3+c] * points[b][c][n]
// WMMA version: one wave per 16 points; M=32 (2 tiles), K=32 (8 tiles),
// B fragments loaded directly from points (coalesced across lanes).
// ---------------------------------------------------------------------------
__global__ __launch_bounds__(128) void feat_kernel(const float* __restrict__ points,
                                                   const float* __restrict__ W0,
                                                   float* __restrict__ feat) {
  __shared__ float sWf[M0_][D_];
  const int tid = threadIdx.x;
  for (int i = tid; i < M0_ * D_; i += 128)
    sWf[i / D_][i % D_] = W0[(i / D_) * W0COLS + 3 + (i % D_)];
  __syncthreads();

  const int w = tid >> 5, lane = tid & 31, half = lane >> 4, m = lane & 15;
  const int gbase = blockIdx.x * 64 + w * 16;   // flat (b, n) tile base
  const int b = gbase / N_, n0 = gbase % N_;
  const float* P = points + (size_t)b * D_ * N_;
  const int n = n0 + m;

  v8f acc0 = {}, acc1 = {};
#pragma unroll
  for (int kt = 0; kt < 8; ++kt) {
    const int kk = kt * 4 + half * 2;
    v2f bb; bb.x = P[(size_t)kk * N_ + n]; bb.y = P[(size_t)(kk + 1) * N_ + n];
    v2f a0; a0.x = sWf[m][kk];      a0.y = sWf[m][kk + 1];
    v2f a1; a1.x = sWf[16 + m][kk]; a1.y = sWf[16 + m][kk + 1];
    acc0 = wmma_f32_4(a0, bb, acc0);
    acc1 = wmma_f32_4(a1, bb, acc1);
  }

  float* F = feat + (size_t)(gbase + m) * M0_;
#pragma unroll
  for (int r = 0; r < 8; ++r) {
    F[r + half * 8]      = acc0[r];
    F[16 + r + half * 8] = acc1[r];
  }
}

// ---------------------------------------------------------------------------
// Kernel B: KNN via f32 WMMA distance tiles + register top-16 per query.
// One wave owns 32 queries; candidates stream 32 per iteration (2 B tiles,
// 4 WMMAs per barrier pair; A fragments reused across both B tiles).
// dist = |q|^2 + |c|^2 - 2 q.c
// ---------------------------------------------------------------------------
__global__ __launch_bounds__(32) void knn_kernel(const float* __restrict__ xyz,
                                                 int* __restrict__ knn_out) {
  __shared__ float csq_sh[32];
  __shared__ float dot_sh[32][33];   // [query][cand 0..31] with pad

  const int lane = threadIdx.x;
  const int half = lane >> 4;
  const int m    = lane & 15;

  const int nblk = N_ / 32;
  const int b  = blockIdx.x / nblk;
  const int qb = (blockIdx.x % nblk) * 32;

  const float* X = xyz + (size_t)b * 3 * N_;
  const float* Y = X + N_;
  const float* Z = X + 2 * N_;

  // A fragments: two 16x4 query tiles. VGPR0 = K{0|2}, VGPR1 = K{1|3}.
  const int q0 = qb + m, q1 = qb + 16 + m;
  v2f a0, a1;
  a0.x = half ? Z[q0] : X[q0];
  a0.y = half ? 0.0f  : Y[q0];
  a1.x = half ? Z[q1] : X[q1];
  a1.y = half ? 0.0f  : Y[q1];

  const float qx = X[qb + lane], qy = Y[qb + lane], qz = Z[qb + lane];
  const float qsq = qx * qx + qy * qy + qz * qz;

  float bestd[KNN_];
  int   besti[KNN_];
#pragma unroll
  for (int t = 0; t < KNN_; ++t) { bestd[t] = 3.0e38f; besti[t] = 0; }
  float worst = 3.0e38f;
  int   wslot = 0;

  for (int c0 = 0; c0 < N_; c0 += 32) {
    const int ca = c0 + m;        // candidate tile 0
    const int cb = c0 + 16 + m;   // candidate tile 1
    const float ax = X[ca], ay = Y[ca], az = Z[ca];
    const float bx = X[cb], by = Y[cb], bz = Z[cb];

    v2f b0; b0.x = half ? az : ax; b0.y = half ? 0.0f : ay;
    v2f b1; b1.x = half ? bz : bx; b1.y = half ? 0.0f : by;

    // lane's "own" candidate is c0 + lane: tile0 for lanes 0-15, tile1 else
    csq_sh[lane] = half ? (bx * bx + by * by + bz * bz)
                        : (ax * ax + ay * ay + az * az);

    v8f zc = {};
    v8f d00 = wmma_f32_4(a0, b0, zc);
    v8f d01 = wmma_f32_4(a0, b1, zc);
    v8f d10 = wmma_f32_4(a1, b0, zc);
    v8f d11 = wmma_f32_4(a1, b1, zc);

    // C layout: lanes 0-15 -> (M=r, N=lane); lanes 16-31 -> (M=r+8, N=lane-16)
#pragma unroll
    for (int r = 0; r < 8; ++r) {
      const int rr = r + half * 8;
      dot_sh[rr][m]           = d00[r];
      dot_sh[rr][16 + m]      = d01[r];
      dot_sh[16 + rr][m]      = d10[r];
      dot_sh[16 + rr][16 + m] = d11[r];
    }
    __syncthreads();

#pragma unroll
    for (int j = 0; j < 32; ++j) {
      const float d = qsq + csq_sh[j] - 2.0f * dot_sh[lane][j];
      const int ci = c0 + j;
      if (d < worst) {
#pragma unroll
        for (int t = 0; t < KNN_; ++t)
          if (t == wslot) { bestd[t] = d; besti[t] = ci; }
        worst = bestd[0]; wslot = 0;
#pragma unroll
        for (int t = 1; t < KNN_; ++t)
          if (bestd[t] > worst) { worst = bestd[t]; wslot = t; }
      }
    }
    __syncthreads();
  }

  int* O = knn_out + ((size_t)b * N_ + qb + lane) * KNN_;
#pragma unroll
  for (int t = 0; t < KNN_; ++t) O[t] = besti[t];
}

// ---------------------------------------------------------------------------
// Kernel C: gather + conv0(W0xyz.rel + feat) + conv1(W1) + conv2(W2) + max_k.
// 4 waves per block, TWO points per wave (A weight fragments reused across
// both points' B tiles). All GEMMs are f32 16x16x4 WMMA; layer0 feeds the
// gathered neighbor features through the C operand.
// ---------------------------------------------------------------------------
__global__ __launch_bounds__(128) void group_conv_kernel(
    const float* __restrict__ xyz, const float* __restrict__ feat,
    const int* __restrict__ knn_idx, const float* __restrict__ W0,
    const float* __restrict__ W1, const float* __restrict__ W2,
    float* __restrict__ out) {
  __shared__ float sW0[M0_][4];              // xyz part of W0, K-padded to 4
  __shared__ float sW1[M1_][M0_];
  __shared__ float sW2[M2_][M1_];
  __shared__ int   idxsh[4][2][16];
  __shared__ float featg[4][2][16][M0_ + 1]; // [p][k][channel]
  __shared__ float xsh[4][M0_][33];          // activations [channel][p*16+k]
  __shared__ float x2sh[4][M2_][33];

  const int tid = threadIdx.x;
  for (int i = tid; i < M0_ * 4; i += 128) {
    int o = i >> 2, c = i & 3;
    sW0[o][c] = (c < 3) ? W0[o * W0COLS + c] : 0.0f;
  }
  for (int i = tid; i < M1_ * M0_; i += 128) sW1[i / M0_][i % M0_] = W1[i];
  for (int i = tid; i < M2_ * M1_; i += 128) sW2[i / M1_][i % M1_] = W2[i];
  __syncthreads();

  const int w    = tid >> 5;
  const int lane = tid & 31;
  const int half = lane >> 4;
  const int m    = lane & 15;

  const int gp0 = (blockIdx.x * 4 + w) * 2;  // first flat (b, point) of pair
  const int b   = gp0 / N_;
  const int pt0 = gp0 % N_;                  // pair stays inside one batch
  const float* X = xyz + (size_t)b * 3 * N_;
  const float* Y = X + N_;
  const float* Z = X + 2 * N_;

  // lane (half, m) loads neighbor index k=m of point p=half
  idxsh[w][half][m] = knn_idx[(size_t)(gp0 + half) * KNN_ + m];
  __syncthreads();

  // Gather neighbor features into LDS: featg[w][p][k][ch]
  for (int j = 0; j < 32; ++j) {
    const int p = j >> 4, k = j & 15;
    const int nk = idxsh[w][p][k];
    featg[w][p][k][lane] = feat[((size_t)b * N_ + nk) * M0_ + lane];
  }
  __syncthreads();

  // Relative coords (B fragments, K padded) for neighbor column m of each point
  v2f rb[2];
#pragma unroll
  for (int p = 0; p < 2; ++p) {
    const int ni = idxsh[w][p][m];
    const int pt = pt0 + p;
    const float rx = X[ni] - X[pt];
    const float ry = Y[ni] - Y[pt];
    const float rz = Z[ni] - Z[pt];
    rb[p].x = half ? rz : rx;
    rb[p].y = half ? 0.0f : ry;
  }

  // Layer 0: x0 = W0[:, :3] @ rel + gathered feat (feat enters through C)
  v8f acc0[2][2];
#pragma unroll
  for (int mt = 0; mt < 2; ++mt) {
    const int row = mt * 16 + m;
    v2f aa;
    aa.x = half ? sW0[row][2] : sW0[row][0];
    aa.y = half ? sW0[row][3] : sW0[row][1];
#pragma unroll
    for (int p = 0; p < 2; ++p) {
      v8f cc;
#pragma unroll
      for (int r = 0; r < 8; ++r) cc[r] = featg[w][p][m][mt * 16 + r + half * 8];
      acc0[mt][p] = wmma_f32_4(aa, rb[p], cc);
    }
  }
  __syncthreads();
#pragma unroll
  for (int mt = 0; mt < 2; ++mt)
#pragma unroll
    for (int p = 0; p < 2; ++p)
#pragma unroll
      for (int r = 0; r < 8; ++r)
        xsh[w][mt * 16 + r + half * 8][p * 16 + m] = lrelu(acc0[mt][p][r]);
  __syncthreads();

  // Layer 1: x1 = W1 @ x0  (2 M-tiles x 2 points x 8 K-tiles)
  v8f acc1[2][2];
#pragma unroll
  for (int mt = 0; mt < 2; ++mt) {
#pragma unroll
    for (int p = 0; p < 2; ++p) acc1[mt][p] = (v8f){};
#pragma unroll
    for (int kt = 0; kt < 8; ++kt) {
      const int kk = kt * 4 + half * 2;
      const int row = mt * 16 + m;
      v2f aa; aa.x = sW1[row][kk]; aa.y = sW1[row][kk + 1];
#pragma unroll
      for (int p = 0; p < 2; ++p) {
        v2f bb;
        bb.x = xsh[w][kk][p * 16 + m];
        bb.y = xsh[w][kk + 1][p * 16 + m];
        acc1[mt][p] = wmma_f32_4(aa, bb, acc1[mt][p]);
      }
    }
  }
  __syncthreads();
#pragma unroll
  for (int mt = 0; mt < 2; ++mt)
#pragma unroll
    for (int p = 0; p < 2; ++p)
#pragma unroll
      for (int r = 0; r < 8; ++r)
        xsh[w][mt * 16 + r + half * 8][p * 16 + m] = lrelu(acc1[mt][p][r]);
  __syncthreads();

  // Layer 2: x2 = W2 @ x1  (4 M-tiles x 2 points x 8 K-tiles)
#pragma unroll
  for (int mt = 0; mt < 4; ++mt) {
    v8f acc2[2];
#pragma unroll
    for (int p = 0; p < 2; ++p) acc2[p] = (v8f){};
#pragma unroll
    for (int kt = 0; kt < 8; ++kt) {
      const int kk = kt * 4 + half * 2;
      const int row = mt * 16 + m;
      v2f aa; aa.x = sW2[row][kk]; aa.y = sW2[row][kk + 1];
#pragma unroll
      for (int p = 0; p < 2; ++p) {
        v2f bb;
        bb.x = xsh[w][kk][p * 16 + m];
        bb.y = xsh[w][kk + 1][p * 16 + m];
        acc2[p] = wmma_f32_4(aa, bb, acc2[p]);
      }
    }
#pragma unroll
    for (int p = 0; p < 2; ++p)
#pragma unroll
      for (int r = 0; r < 8; ++r)
        x2sh[w][mt * 16 + r + half * 8][p * 16 + m] = lrelu(acc2[p][r]);
  }
  __syncthreads();

  // Max over the 16 neighbor columns of each point; 4 outputs per lane.
#pragma unroll
  for (int p = 0; p < 2; ++p) {
#pragma unroll
    for (int h = 0; h < 2; ++h) {
      const int o = lane + h * 32;
      float mx = x2sh[w][o][p * 16];
#pragma unroll
      for (int j = 1; j < 16; ++j) mx = fmaxf(mx, x2sh[w][o][p * 16 + j]);
      out[((size_t)b * M2_ + o) * N_ + (pt0 + p)] = mx;
    }
  }
}

// ---------------------------------------------------------------------------
extern "C" void kernel_launch(void* const* d_in, const int* in_sizes, int n_in,
                              void* d_out, int out_size, void* d_ws, size_t ws_size,
                              hipStream_t stream) {
  (void)in_sizes; (void)n_in; (void)out_size; (void)ws_size;
  const float* xyz    = (const float*)d_in[0];   // [B,3,N]
  const float* points = (const float*)d_in[1];   // [B,32,N]
  const float* W0     = (const float*)d_in[2];   // [32,35]
  const float* W1     = (const float*)d_in[3];   // [32,32]
  const float* W2     = (const float*)d_in[4];   // [64,32]
  float* out = (float*)d_out;                    // [B,64,N]

  float* feat = (float*)d_ws;                                   // B*N*32 f32
  int*   knn  = (int*)((char*)d_ws +
                       (size_t)B_ * N_ * M0_ * sizeof(float));  // B*N*16 i32

  feat_kernel<<<B_ * N_ / 64, 128, 0, stream>>>(points, W0, feat);
  knn_kernel<<<B_ * (N_ / 32), 32, 0, stream>>>(xyz, knn);
  group_conv_kernel<<<B_ * N_ / 8, 128, 0, stream>>>(xyz, feat, knn, W0, W1, W2, out);
}